// WordGCNPool_23235773072063
// MI455X (gfx1250) — compile-verified
//
#include <hip/hip_runtime.h>

typedef __attribute__((ext_vector_type(2))) float v2f;
typedef __attribute__((ext_vector_type(8))) float v8f;

#define TPB  256
#define LDSP 132   // padded LDS row stride (dwords) for W tile

// ---------------- zero fill ----------------
__global__ __launch_bounds__(TPB)
void k_zero(float* __restrict__ p, long n) {
  long i = (long)blockIdx.x * TPB + threadIdx.x;
  long stride = (long)gridDim.x * TPB;
  for (; i < n; i += stride) p[i] = 0.0f;
}

// ---------------- SpMM: out[row[e], :] += val[e] * dense[col[e], :] ----------------
// one wave per edge (grid-stride); lane covers cols {l, l+32, l+64, l+96}
// unsafeAtomicAdd -> native global_atomic_add_f32 (L2-side RMW, no CAS loop)
__global__ __launch_bounds__(TPB)
void k_spmm(const int* __restrict__ row, const int* __restrict__ col,
            const float* __restrict__ val, const float* __restrict__ dense,
            float* __restrict__ out, int nnz) {
  int lane   = threadIdx.x & 31;
  int wave   = blockIdx.x * (TPB / 32) + (threadIdx.x >> 5);
  int nwaves = gridDim.x * (TPB / 32);
  for (int e = wave; e < nnz; e += nwaves) {
    int r = row[e], c = col[e];
    float v = val[e];
    const float* src = dense + (size_t)c * 128 + lane;
    float*       dst = out   + (size_t)r * 128 + lane;
#pragma unroll
    for (int j = 0; j < 4; ++j)
      unsafeAtomicAdd(dst + 32 * j, v * src[32 * j]);
  }
}

// ---------------- GEMM: OUT[n_rows,128] = act(X[n_rows,128] @ W[128,128]^T + bias) ----------------
// Block = 8 waves; wave handles a 16-row strip, all 128 output cols (8 x v8f acc).
// WMMA f32 16x16x4, W staged in LDS.
__global__ __launch_bounds__(TPB)
void k_gemm128(const float* __restrict__ X, const float* __restrict__ W,
               const float* __restrict__ bias, float* __restrict__ OUT,
               int n_rows, int relu) {
  __shared__ float wlds[128 * LDSP];
  int tid = threadIdx.x;
#pragma unroll 4
  for (int i = tid; i < 128 * 128; i += TPB)
    wlds[(i >> 7) * LDSP + (i & 127)] = W[i];
  __syncthreads();

  int lane  = tid & 31;
  int mlane = lane & 15;          // M (for A) / N (for B, C/D) within tile
  int khalf = (lane >> 4) << 1;   // lanes 16..31 hold K+2,K+3
  int m0    = blockIdx.x * 128 + (tid >> 5) * 16;

  int arow = m0 + mlane;
  if (arow >= n_rows) arow = 0;   // clamp: garbage rows never stored
  const float* aptr = X + (size_t)arow * 128 + khalf;

  v8f acc[8] = {};
  for (int k = 0; k < 128; k += 4) {
    v2f a = *(const v2f*)(aptr + k);                       // X[arow, k+khalf .. +1]
#pragma unroll
    for (int nt = 0; nt < 8; ++nt) {
      // B[kk][n] = W[n][kk]; lane holds W[nt*16+mlane, k+khalf .. +1]
      v2f b = *(const v2f*)(&wlds[(nt * 16 + mlane) * LDSP + k + khalf]);
      acc[nt] = __builtin_amdgcn_wmma_f32_16x16x4_f32(
          false, a, false, b, (short)0, acc[nt], false, false);
    }
  }

  // D layout: vgpr j -> M = m0 + j + 8*(lane>>4), N = nt*16 + (lane&15)
  int mo = m0 + 8 * (lane >> 4);
#pragma unroll
  for (int j = 0; j < 8; ++j) {
    int m = mo + j;
    if (m < n_rows) {
#pragma unroll
      for (int nt = 0; nt < 8; ++nt) {
        int c = nt * 16 + mlane;
        float v = acc[nt][j];
        if (bias) v += bias[c];
        if (relu) v = fmaxf(v, 0.0f);
        OUT[(size_t)m * 128 + c] = v;
      }
    }
  }
}

// ---------------- residual + LayerNorm + add-emb (fused pooling input) ----------------
// out[r,:] = LN(0.3*emb[r,:] + 0.7*act[r,:]) * g + b + emb[r,:]
__global__ __launch_bounds__(TPB)
void k_resid_ln(const float* __restrict__ act, const float* __restrict__ emb,
                const float* __restrict__ g, const float* __restrict__ b,
                float* __restrict__ out, int nrows) {
  int r = blockIdx.x * (TPB / 32) + (threadIdx.x >> 5);
  if (r >= nrows) return;
  int lane = threadIdx.x & 31;
  float h[4], e[4];
  float s = 0.0f, sq = 0.0f;
#pragma unroll
  for (int j = 0; j < 4; ++j) {
    int c = lane + 32 * j;
    e[j] = emb[(size_t)r * 128 + c];
    h[j] = 0.3f * e[j] + 0.7f * act[(size_t)r * 128 + c];
    s  += h[j];
    sq += h[j] * h[j];
  }
#pragma unroll
  for (int off = 16; off > 0; off >>= 1) {
    s  += __shfl_xor(s,  off, 32);
    sq += __shfl_xor(sq, off, 32);
  }
  float mu  = s * (1.0f / 128.0f);
  float var = sq * (1.0f / 128.0f) - mu * mu;
  float rs  = rsqrtf(var + 1e-5f);
#pragma unroll
  for (int j = 0; j < 4; ++j) {
    int c = lane + 32 * j;
    out[(size_t)r * 128 + c] = (h[j] - mu) * rs * g[c] + b[c] + e[j];
  }
}

// ---------------- classifier: logits[d, 0..1] = doc[d,:] @ clfW[0..1,:] + clfb ----------------
__global__ __launch_bounds__(TPB)
void k_clf(const float* __restrict__ doc, const float* __restrict__ clfW,
           const float* __restrict__ clfb, float* __restrict__ out, int ndocs) {
  int d = blockIdx.x * (TPB / 32) + (threadIdx.x >> 5);
  if (d >= ndocs) return;
  int lane = threadIdx.x & 31;
  float d0 = 0.0f, d1 = 0.0f;
#pragma unroll
  for (int j = 0; j < 4; ++j) {
    int c = lane + 32 * j;
    float x = doc[(size_t)d * 128 + c];
    d0 += x * clfW[c];
    d1 += x * clfW[128 + c];
  }
#pragma unroll
  for (int off = 16; off > 0; off >>= 1) {
    d0 += __shfl_xor(d0, off, 32);
    d1 += __shfl_xor(d1, off, 32);
  }
  if (lane == 0) {
    out[(size_t)d * 2 + 0] = d0 + clfb[0];
    out[(size_t)d * 2 + 1] = d1 + clfb[1];
  }
}

// ---------------- driver ----------------
extern "C" void kernel_launch(void* const* d_in, const int* in_sizes, int n_in,
                              void* d_out, int out_size, void* d_ws, size_t ws_size,
                              hipStream_t stream) {
  (void)n_in; (void)ws_size;
  const int*   A_row  = (const int*)  d_in[0];
  const int*   A_col  = (const int*)  d_in[1];
  const float* A_val  = (const float*)d_in[2];
  const int*   X_row  = (const int*)  d_in[3];
  const int*   X_col  = (const int*)  d_in[4];
  const float* X_val  = (const float*)d_in[5];
  const float* emb_W  = (const float*)d_in[6];
  const float* lin1_W = (const float*)d_in[7];
  const float* lin2_W = (const float*)d_in[8];
  const float* norm_g = (const float*)d_in[9];
  const float* norm_b = (const float*)d_in[10];
  const float* mlp_W  = (const float*)d_in[11];
  const float* mlp_b  = (const float*)d_in[12];
  const float* clf_W  = (const float*)d_in[13];
  const float* clf_b  = (const float*)d_in[14];
  float* logits = (float*)d_out;

  const int E     = in_sizes[0];
  const int NNZ   = in_sizes[3];
  const int V     = in_sizes[6] / 128;
  const int NDOCS = out_size / 2;

  float* B1 = (float*)d_ws;                 // [V,128]
  float* B2 = B1 + (size_t)V * 128;         // [V,128]
  float* D1 = B2 + (size_t)V * 128;         // [NDOCS,128]

  const long nV = (long)V * 128;
  const long nD = (long)NDOCS * 128;
  dim3 blk(TPB);
  const int zgrid = 1024;
  const int sgrid = 4096;
  const int gV = (V + 127) / 128;
  const int gD = (NDOCS + 127) / 128;
  const int wV = (V + 7) / 8;
  const int wD = (NDOCS + 7) / 8;

  // GCN layer 1: B1 = A @ emb ; B2 = relu(B1 @ lin1^T)
  k_zero   <<<zgrid, blk, 0, stream>>>(B1, nV);
  k_spmm   <<<sgrid, blk, 0, stream>>>(A_row, A_col, A_val, emb_W, B1, E);
  k_gemm128<<<gV,    blk, 0, stream>>>(B1, lin1_W, nullptr, B2, V, 1);

  // GCN layer 2: B1 = A @ B2 ; B2 = relu(B1 @ lin2^T)
  k_zero   <<<zgrid, blk, 0, stream>>>(B1, nV);
  k_spmm   <<<sgrid, blk, 0, stream>>>(A_row, A_col, A_val, B2, B1, E);
  k_gemm128<<<gV,    blk, 0, stream>>>(B1, lin2_W, nullptr, B2, V, 1);

  // B1 = LN(0.3*emb + 0.7*B2)*g + b + emb   (word_H + emb, fused pooling operand)
  k_resid_ln<<<wV, blk, 0, stream>>>(B2, emb_W, norm_g, norm_b, B1, V);

  // doc pooling: D1 = X @ B1  (== X@word_H + X@emb by linearity)
  k_zero   <<<zgrid, blk, 0, stream>>>(D1, nD);
  k_spmm   <<<sgrid, blk, 0, stream>>>(X_row, X_col, X_val, B1, D1, NNZ);

  // MLP head (in place) + classifier
  k_gemm128<<<gD, blk, 0, stream>>>(D1, mlp_W, mlp_b, D1, NDOCS, 1);
  k_clf    <<<wD, blk, 0, stream>>>(D1, clf_W, clf_b, logits, NDOCS);
}